// TransLayer_73873437491683
// MI455X (gfx1250) — compile-verified
//
#include <hip/hip_runtime.h>

typedef __attribute__((ext_vector_type(2))) float v2f;
typedef __attribute__((ext_vector_type(8))) float v8f;

#define TM 128
#define TN 64
#define TK 32
#define ASTR 36   // LDS row stride: conflict-free b64 frag reads, 16B-aligned rows

// ---------------------------------------------------------------------------
// Generic batched GEMM, fp32 WMMA (V_WMMA_F32_16X16X4_F32).
//   D = alpha * A(MxK) @ op(B)(KxN) + beta * C + bias
// Batch offset of each matrix: (bat>>3)*s0 + (bat&7)*s1  (uniform stride S ->
// s0=8S, s1=S). tb==1 means B stored (N x K) row-major (i.e. use B^T).
// M multiple of 128, N multiple of 64, K multiple of 32 in this pipeline.
// Each of 8 waves owns a 32x32 C tile: 4 wmma per 4 frag loads (A/B reuse x2).
// ---------------------------------------------------------------------------
__global__ __launch_bounds__(256) void gemm_wmma_f32(
    const float* __restrict__ A, int lda, long long aS0, long long aS1,
    const float* __restrict__ B, int ldb, long long bS0, long long bS1, int tb,
    const float* __restrict__ Cp, int ldc, long long cS0, long long cS1, float beta,
    float* __restrict__ D, int ldd, long long dS0, long long dS1,
    int M, int N, int K, float alpha, const float* __restrict__ bias) {
  (void)M; (void)N;
  __shared__ float As[TM][ASTR];   // [m][k]
  __shared__ float Bs[TN][ASTR];   // [n][k]

  const int tid  = threadIdx.x;
  const int lane = tid & 31;       // wave32
  const int wid  = tid >> 5;       // 8 waves
  const int bn0  = blockIdx.x * TN;
  const int bm0  = blockIdx.y * TM;
  const int bat  = blockIdx.z;

  const float* Ab = A + ((long long)(bat >> 3) * aS0 + (long long)(bat & 7) * aS1);
  const float* Bb = B + ((long long)(bat >> 3) * bS0 + (long long)(bat & 7) * bS1);
  const float* Cb = Cp ? Cp + ((long long)(bat >> 3) * cS0 + (long long)(bat & 7) * cS1) : nullptr;
  float*       Db = D + ((long long)(bat >> 3) * dS0 + (long long)(bat & 7) * dS1);

  const int wm = (wid >> 1) * 32;  // wave M offset in tile (4 waves over M)
  const int wn = (wid & 1) * 32;   // wave N offset in tile (2 waves over N)

  // ISA fragment mapping (32-bit A 16x4 / B 4x16):
  //   lanes 0-15: row/col = lane,    K = {0,1} in regs {0,1}
  //   lanes16-31: row/col = lane-16, K = {2,3} in regs {0,1}
  const int mfr   = lane & 15;
  const int khalf = (lane >> 4) * 2;

  v8f c00 = {0.f,0.f,0.f,0.f,0.f,0.f,0.f,0.f};
  v8f c01 = c00, c10 = c00, c11 = c00;

  for (int k0 = 0; k0 < K; k0 += TK) {
    // ---- stage A tile (128 x 32): float4 over k, 4 passes of 32 rows
    {
      int kk = (tid & 7) << 2;
      int mm = tid >> 3;
#pragma unroll
      for (int p = 0; p < 4; ++p, mm += 32) {
        float4 v = *(const float4*)(Ab + (long long)(bm0 + mm) * lda + (k0 + kk));
        *(float4*)&As[mm][kk] = v;
      }
      if (k0 + TK < K)
        __builtin_prefetch((const void*)(Ab + (long long)(bm0 + (tid >> 3)) * lda + (k0 + TK)), 0, 0);
    }
    // ---- stage B tile into Bs[n][k]
    if (tb) {  // B is (N x K) row-major: float4 over k
      int kk = (tid & 7) << 2;
      int nn = tid >> 3;
#pragma unroll
      for (int p = 0; p < 2; ++p, nn += 32) {
        float4 v = *(const float4*)(Bb + (long long)(bn0 + nn) * ldb + (k0 + kk));
        *(float4*)&Bs[nn][kk] = v;
      }
      if (k0 + TK < K)
        __builtin_prefetch((const void*)(Bb + (long long)(bn0 + (tid >> 3)) * ldb + (k0 + TK)), 0, 0);
    } else {   // B is (K x N) row-major: float4 over n, transpose into LDS
      int nn4 = (tid & 15) << 2;
      int kk  = tid >> 4;
#pragma unroll
      for (int p = 0; p < 2; ++p, kk += 16) {
        float4 v = *(const float4*)(Bb + (long long)(k0 + kk) * ldb + (bn0 + nn4));
        Bs[nn4 + 0][kk] = v.x;
        Bs[nn4 + 1][kk] = v.y;
        Bs[nn4 + 2][kk] = v.z;
        Bs[nn4 + 3][kk] = v.w;
      }
      if (k0 + TK < K)
        __builtin_prefetch((const void*)(Bb + (long long)(k0 + TK + (tid >> 4)) * ldb + bn0), 0, 0);
    }
    __syncthreads();

#pragma unroll
    for (int ks = 0; ks < TK; ks += 4) {
      v2f a0 = *(const v2f*)&As[wm + mfr][ks + khalf];
      v2f a1 = *(const v2f*)&As[wm + 16 + mfr][ks + khalf];
      v2f b0 = *(const v2f*)&Bs[wn + mfr][ks + khalf];
      v2f b1 = *(const v2f*)&Bs[wn + 16 + mfr][ks + khalf];
      c00 = __builtin_amdgcn_wmma_f32_16x16x4_f32(false, a0, false, b0, (short)0, c00, false, false);
      c01 = __builtin_amdgcn_wmma_f32_16x16x4_f32(false, a0, false, b1, (short)0, c01, false, false);
      c10 = __builtin_amdgcn_wmma_f32_16x16x4_f32(false, a1, false, b0, (short)0, c10, false, false);
      c11 = __builtin_amdgcn_wmma_f32_16x16x4_f32(false, a1, false, b1, (short)0, c11, false, false);
    }
    __syncthreads();
  }

  // ---- epilogue. C/D layout: reg r -> M = r + 8*(lane>>4), N = lane&15
  const int colb = lane & 15;
  const int rowb = (lane >> 4) << 3;
#pragma unroll
  for (int fm = 0; fm < 2; ++fm) {
#pragma unroll
    for (int fn = 0; fn < 2; ++fn) {
      v8f acc = fm ? (fn ? c11 : c10) : (fn ? c01 : c00);
      const int ncol = bn0 + wn + fn * 16 + colb;
      float bv = bias ? bias[ncol] : 0.f;
#pragma unroll
      for (int r = 0; r < 8; ++r) {
        const int row = bm0 + wm + fm * 16 + rowb + r;
        float val = alpha * acc[r] + bv;
        if (Cb) val += beta * Cb[(long long)row * ldc + ncol];
        Db[(long long)row * ldd + ncol] = val;
      }
    }
  }
}

// ---------------------------------------------------------------------------
// LayerNorm over dim=512, one block per row.
// ---------------------------------------------------------------------------
__global__ __launch_bounds__(256) void ln_kernel(
    const float* __restrict__ x, const float* __restrict__ gamma,
    const float* __restrict__ beta, float* __restrict__ xn) {
  __shared__ float red[256];
  const int tid = threadIdx.x;
  const long long row = blockIdx.x;
  const float* xr = x + row * 512;
  float v0 = xr[tid], v1 = xr[tid + 256];
  red[tid] = v0 + v1;
  __syncthreads();
  for (int o = 128; o > 0; o >>= 1) { if (tid < o) red[tid] += red[tid + o]; __syncthreads(); }
  float mu = red[0] * (1.0f / 512.0f);
  __syncthreads();
  float d0 = v0 - mu, d1 = v1 - mu;
  red[tid] = d0 * d0 + d1 * d1;
  __syncthreads();
  for (int o = 128; o > 0; o >>= 1) { if (tid < o) red[tid] += red[tid + o]; __syncthreads(); }
  float inv = rsqrtf(red[0] * (1.0f / 512.0f) + 1e-5f);
  float* orow = xn + row * 512;
  orow[tid]       = d0 * inv * gamma[tid]       + beta[tid];
  orow[tid + 256] = d1 * inv * gamma[tid + 256] + beta[tid + 256];
}

// ---------------------------------------------------------------------------
// In-place row softmax, one block per row.
// ---------------------------------------------------------------------------
__global__ __launch_bounds__(256) void softmax_kernel(float* __restrict__ p, int cols) {
  __shared__ float red[256];
  const int tid = threadIdx.x;
  float* r = p + (long long)blockIdx.x * cols;
  float mx = -3.4e38f;
  for (int j = tid; j < cols; j += 256) mx = fmaxf(mx, r[j]);
  red[tid] = mx; __syncthreads();
  for (int o = 128; o > 0; o >>= 1) { if (tid < o) red[tid] = fmaxf(red[tid], red[tid + o]); __syncthreads(); }
  mx = red[0]; __syncthreads();
  float s = 0.f;
  for (int j = tid; j < cols; j += 256) { float e = __expf(r[j] - mx); r[j] = e; s += e; }
  red[tid] = s; __syncthreads();
  for (int o = 128; o > 0; o >>= 1) { if (tid < o) red[tid] += red[tid + o]; __syncthreads(); }
  float inv = 1.0f / red[0];
  for (int j = tid; j < cols; j += 256) r[j] *= inv;
}

// mean over 16 consecutive tokens: (bh,4096,64) -> (bh,256,64)
__global__ __launch_bounds__(256) void pool16(const float* __restrict__ in, float* __restrict__ out) {
  const int idx = blockIdx.x * 256 + threadIdx.x;        // bh*16384 + m*64 + d
  const int d  = idx & 63;
  const int mI = (idx >> 6) & 255;
  const int bh = idx >> 14;
  const float* p = in + ((long long)bh * 4096 + mI * 16) * 64 + d;
  float s = 0.f;
#pragma unroll
  for (int j = 0; j < 16; ++j) s += p[j * 64];
  out[idx] = s * (1.0f / 16.0f);
}

__global__ void init2(float* s) { if (threadIdx.x < 2) s[threadIdx.x] = 0.f; }

__device__ __forceinline__ void atomicMaxPosF(float* p, float v) {
  atomicMax((int*)p, __float_as_int(v));  // valid for non-negative floats
}

// col = max over rows of row-sum(|a2|); one block per row of 256
__global__ __launch_bounds__(256) void rowsum_max(const float* __restrict__ a2, float* __restrict__ scal) {
  __shared__ float red[256];
  const int tid = threadIdx.x;
  red[tid] = fabsf(a2[(long long)blockIdx.x * 256 + tid]);
  __syncthreads();
  for (int o = 128; o > 0; o >>= 1) { if (tid < o) red[tid] += red[tid + o]; __syncthreads(); }
  if (tid == 0) atomicMaxPosF(&scal[0], red[0]);
}

// row = max over cols of col-sum(|a2|); one block per bh, thread per column
__global__ __launch_bounds__(256) void colsum_max(const float* __restrict__ a2, float* __restrict__ scal) {
  const float* p = a2 + (long long)blockIdx.x * 65536 + threadIdx.x;
  float s = 0.f;
  for (int i = 0; i < 256; ++i) s += fabsf(p[i * 256]);
  atomicMaxPosF(&scal[1], s);
}

// z0 = a2^T / (col*row)
__global__ __launch_bounds__(256) void transpose_scale(const float* __restrict__ a2,
                                                       const float* __restrict__ scal,
                                                       float* __restrict__ z) {
  const int idx = blockIdx.x * 256 + threadIdx.x;        // bh*65536 + i*256 + j
  const int j  = idx & 255;
  const int i  = (idx >> 8) & 255;
  const int bh = idx >> 16;
  const float inv = 1.0f / (scal[0] * scal[1]);
  z[idx] = a2[(long long)bh * 65536 + j * 256 + i] * inv;
}

// depthwise conv over sequence dim, kernel 33, zero pad 16, weight per head
__global__ __launch_bounds__(256) void dwconv33(const float* __restrict__ V,
                                                const float* __restrict__ w,
                                                float* __restrict__ out) {
  __shared__ float wl[33];
  const int tid = threadIdx.x;
  const int bh = blockIdx.y;
  if (tid < 33) wl[tid] = w[(bh & 7) * 33 + tid];
  __syncthreads();
  const int d = tid & 63;
  const int n = blockIdx.x * 4 + (tid >> 6);
  const float* vp = V + (long long)bh * 4096 * 64 + d;
  float s = 0.f;
#pragma unroll
  for (int t = 0; t < 33; ++t) {
    int nn = n + t - 16;
    if (nn >= 0 && nn < 4096) s += wl[t] * vp[(long long)nn * 64];
  }
  out[((long long)bh * 4096 + n) * 64 + d] = s;
}

// ---------------------------------------------------------------------------
extern "C" void kernel_launch(void* const* d_in, const int* in_sizes, int n_in,
                              void* d_out, int out_size, void* d_ws, size_t ws_size,
                              hipStream_t stream) {
  (void)in_sizes; (void)n_in; (void)out_size; (void)ws_size;
  const float* x     = (const float*)d_in[0];
  const float* gamma = (const float*)d_in[1];
  const float* betap = (const float*)d_in[2];
  const float* Wqkv  = (const float*)d_in[3];
  const float* Wout  = (const float*)d_in[4];
  const float* bout  = (const float*)d_in[5];
  const float* convw = (const float*)d_in[6];
  float* out = (float*)d_out;

  // workspace carve (floats)
  float* W = (float*)d_ws;
  size_t o = 0;
  float* XN  = W + o; o += 16384LL * 512;     // also final pre-proj buffer (b,n,512)
  float* Qb  = W + o; o += 32LL * 4096 * 64;  // also conv residual buffer
  float* Kb  = W + o; o += 32LL * 4096 * 64;
  float* Vb  = W + o; o += 32LL * 4096 * 64;
  float* QL  = W + o; o += 32LL * 256 * 64;
  float* KL  = W + o; o += 32LL * 256 * 64;
  float* A1  = W + o; o += 32LL * 4096 * 256;
  float* A3  = W + o; o += 32LL * 256 * 4096; // also out1 = a1 @ a2inv
  float* A2  = W + o; o += 32LL * 256 * 256;
  float* Zb  = W + o; o += 32LL * 256 * 256;
  float* XZ  = W + o; o += 32LL * 256 * 256;
  float* W2b = W + o; o += 32LL * 256 * 256;
  float* W4b = W + o; o += 32LL * 256 * 256;
  float* ZNb = W + o; o += 32LL * 256 * 256;
  float* A3V = W + o; o += 32LL * 256 * 64;
  float* scal = W + o; o += 2;

  auto G = [&](const float* A, int lda, long long a0, long long a1,
               const float* B, int ldb, long long b0, long long b1, int tb,
               const float* C, int ldc, long long c0, long long c1, float beta,
               float* D, int ldd, long long dd0, long long dd1,
               int M, int N, int K, int batch, float alpha, const float* bias) {
    dim3 g(N / TN, M / TM, batch);
    gemm_wmma_f32<<<g, 256, 0, stream>>>(A, lda, a0, a1, B, ldb, b0, b1, tb,
                                         C, ldc, c0, c1, beta, D, ldd, dd0, dd1,
                                         M, N, K, alpha, bias);
  };
#define US(S) (8LL * (S)), ((long long)(S))  /* uniform per-batch stride */

  // 1. pre-LayerNorm
  ln_kernel<<<16384, 256, 0, stream>>>(x, gamma, betap, XN);

  // 2. QKV projection, head-batched (32 batches), writes (b,h,n,64); q scaled
  const long long Ab0 = 4096LL * 512;  // per-b stride of XN (aS1=0)
  G(XN, 512, Ab0, 0, Wqkv + 0,    1536, 0, 64, 0, nullptr, 0, 0, 0, 0.f, Qb, 64, US(262144), 4096, 64, 512, 32, 0.125f, nullptr);
  G(XN, 512, Ab0, 0, Wqkv + 512,  1536, 0, 64, 0, nullptr, 0, 0, 0, 0.f, Kb, 64, US(262144), 4096, 64, 512, 32, 1.0f,   nullptr);
  G(XN, 512, Ab0, 0, Wqkv + 1024, 1536, 0, 64, 0, nullptr, 0, 0, 0, 0.f, Vb, 64, US(262144), 4096, 64, 512, 32, 1.0f,   nullptr);

  // 3. landmarks (mean over 16 tokens)
  pool16<<<2048, 256, 0, stream>>>(Qb, QL);
  pool16<<<2048, 256, 0, stream>>>(Kb, KL);

  // 4. similarity GEMMs
  G(Qb, 64, US(262144), KL, 64, US(16384),  1, nullptr, 0, 0, 0, 0.f, A1, 256,  US(1048576), 4096, 256,  64, 32, 1.f, nullptr);
  G(QL, 64, US(16384),  KL, 64, US(16384),  1, nullptr, 0, 0, 0, 0.f, A2, 256,  US(65536),    256, 256,  64, 32, 1.f, nullptr);
  G(QL, 64, US(16384),  Kb, 64, US(262144), 1, nullptr, 0, 0, 0, 0.f, A3, 4096, US(1048576),  256, 4096, 64, 32, 1.f, nullptr);

  // 5. softmaxes
  softmax_kernel<<<131072, 256, 0, stream>>>(A1, 256);
  softmax_kernel<<<8192,   256, 0, stream>>>(A2, 256);
  softmax_kernel<<<8192,   256, 0, stream>>>(A3, 4096);

  // 6. pinv init: z = a2^T / (maxRowSum * maxColSum)
  init2<<<1, 32, 0, stream>>>(scal);
  rowsum_max<<<8192, 256, 0, stream>>>(A2, scal);
  colsum_max<<<32,   256, 0, stream>>>(A2, scal);
  transpose_scale<<<8192, 256, 0, stream>>>(A2, scal, Zb);

  // 7. Newton-Schulz: z = 0.25 z (13I - xz(15I - xz(7I - xz)))
  float* z = Zb; float* zn = ZNb;
  for (int it = 0; it < 6; ++it) {
    G(A2, 256, US(65536), z,   256, US(65536), 0, nullptr, 0, 0, 0, 0.f,  XZ,  256, US(65536), 256, 256, 256, 32,  1.0f,  nullptr);
    G(XZ, 256, US(65536), XZ,  256, US(65536), 0, XZ, 256, US(65536), 7.0f,  W2b, 256, US(65536), 256, 256, 256, 32, -1.0f,  nullptr);
    G(XZ, 256, US(65536), W2b, 256, US(65536), 0, XZ, 256, US(65536), 15.0f, W4b, 256, US(65536), 256, 256, 256, 32, -1.0f,  nullptr);
    G(z,  256, US(65536), W4b, 256, US(65536), 0, z,  256, US(65536), 3.25f, zn,  256, US(65536), 256, 256, 256, 32, -0.25f, nullptr);
    float* t = z; z = zn; zn = t;
  }

  // 8. a3v = a3 @ v  (256 x 64, K=4096)
  G(A3, 4096, US(1048576), Vb, 64, US(262144), 0, nullptr, 0, 0, 0, 0.f, A3V, 64, US(16384), 256, 64, 4096, 32, 1.f, nullptr);

  // 9. out1 = a1 @ a2inv  (4096 x 256) -> reuse A3 buffer
  G(A1, 256, US(1048576), z, 256, US(65536), 0, nullptr, 0, 0, 0, 0.f, A3, 256, US(1048576), 4096, 256, 256, 32, 1.f, nullptr);

  // 10. depthwise conv residual of v -> reuse Qb
  dwconv33<<<dim3(1024, 32), 256, 0, stream>>>(Vb, convw, Qb);

  // 11. out2 = out1 @ a3v + conv, scattered into (b,n,512) layout (reuse XN)
  G(A3, 256, US(1048576), A3V, 64, US(16384), 0, Qb, 64, US(262144), 1.0f,
    XN, 512, 2097152LL, 64LL, 4096, 64, 256, 32, 1.f, nullptr);

  // 12. final: out = x + out2 @ W_out + b_out
  G(XN, 512, 0, 0, Wout, 512, 0, 0, 0, x, 512, 0, 0, 1.0f,
    out, 512, 0, 0, 16384, 512, 512, 1, 1.f, bout);
#undef US
}